// BagOfWords_88115549045539
// MI455X (gfx1250) — compile-verified
//
#include <hip/hip_runtime.h>
#include <hip/hip_bf16.h>

// BagOfWords histogram, MI455X (gfx1250).
//
// Reference: counts[b, v] = #{j : tokens[b, j] == v}, return counts[:, 1:].
// B=1024, S=512, V=50257 -> out is [1024, 50256] fp32 (205.8 MB).
//
// Roofline: ~0 FLOPs, ~208 MB of memory traffic -> pure bandwidth problem
// (~9 us floor at 23.3 TB/s). No matrix structure (one-hot GEMM would move
// ~105 GB), so no WMMA: optimal = b128 zero-fill + native fp32 L2 atomics.
//
// NOTE: bow_scatter is defined FIRST so the disasm snippet shows the atomic
// lowering (expect no-return global_atomic_add_f32, not a cmpswap loop).

typedef float v4f __attribute__((ext_vector_type(4)));
typedef int   v4i __attribute__((ext_vector_type(4)));

static constexpr int kB    = 1024;
static constexpr int kS    = 512;
static constexpr int kV    = 50257;
static constexpr int kOutC = kV - 1;  // 50256 (token 0 dropped), divisible by 4

// ---------------------------------------------------------------------------
// Scatter: each thread loads 4 tokens with one global_load_b128 (4 consecutive
// tokens always share a batch row: 512 % 4 == 0) and issues up to 4 no-return
// global_atomic_add_f32 ops. RMWs resolve in L2 (192 MB L2 holds essentially
// the whole output); collisions within a row are ~2.6 expected (birthday
// bound on 512 draws from 50K), so contention is negligible.
// ---------------------------------------------------------------------------
__global__ void __launch_bounds__(256) bow_scatter(const v4i* __restrict__ toks,
                                                   float* __restrict__ out) {
  int t = blockIdx.x * blockDim.x + threadIdx.x;  // quad index, 0 .. B*S/4-1
  v4i v = toks[t];                                 // global_load_b128
  int b = (t << 2) >> 9;                           // batch row (= t*4 / 512)
  float* row = out + (long long)b * kOutC;
#pragma unroll
  for (int k = 0; k < 4; ++k) {
    int tok = v[k];
    if (tok > 0) {
      // Relaxed, device-scope fp32 add, result discarded -> backend picks the
      // no-return GLOBAL_ATOMIC_ADD_F32 (tracked by STOREcnt).
      (void)__hip_atomic_fetch_add(row + (tok - 1), 1.0f,
                                   __ATOMIC_RELAXED, __HIP_MEMORY_SCOPE_AGENT);
    }
  }
}

// ---------------------------------------------------------------------------
// Zero-fill: 128-bit stores, grid-stride, unrolled x4 so the loop is
// store-issue-rate limited rather than loop-overhead limited.
// ---------------------------------------------------------------------------
__global__ void __launch_bounds__(256) bow_zero_fill(v4f* __restrict__ out, int n4) {
  int i      = blockIdx.x * blockDim.x + threadIdx.x;
  int stride = gridDim.x * blockDim.x;
  v4f z = (v4f)0.0f;
  // Main unrolled body: 4 coalesced global_store_b128 per trip.
  for (; i + 3 * stride < n4; i += 4 * stride) {
    out[i]              = z;
    out[i + stride]     = z;
    out[i + 2 * stride] = z;
    out[i + 3 * stride] = z;
  }
  // Tail.
  for (; i < n4; i += stride) {
    out[i] = z;
  }
}

extern "C" void kernel_launch(void* const* d_in, const int* in_sizes, int n_in,
                              void* d_out, int out_size, void* d_ws, size_t ws_size,
                              hipStream_t stream) {
  const v4i* toks = (const v4i*)d_in[0];
  float*     out  = (float*)d_out;

  // out_size = 1024 * 50256 = 51,462,144 floats; divisible by 4.
  int n4 = out_size >> 2;  // 12,865,536 float4 stores

  // Fill: 2048 blocks x 256 threads (wave32 -> 8 waves/block), ~24 b128
  // stores per thread (6 unrolled trips + tail). Enough resident waves to
  // saturate HBM write bandwidth.
  bow_zero_fill<<<2048, 256, 0, stream>>>((v4f*)out, n4);

  // Scatter: one thread per 4 tokens. 1024*512/4 = 131,072 threads.
  int quads = (kB * kS) / 4;
  bow_scatter<<<quads / 256, 256, 0, stream>>>(toks, out);
}